// V1Column_33380485825263
// MI455X (gfx1250) — compile-verified
//
#include <hip/hip_runtime.h>

// ---- Problem constants (match reference) ----
#define NN   50000
#define DD   5
#define NB   5
#define BB   4
#define E_REC 2000000
#define N_LGN 17400
#define E_LGN 600000
#define N_BKG 100
#define E_BKG 200000
static constexpr float DT_C = 1.0f;

// Native fp32 global atomic add (global_atomic_add_f32, no-return -> STOREcnt)
__device__ __forceinline__ void atomic_add_f32(float* p, float v) {
    unsafeAtomicAdd(p, v);
}

// ---- Zero the synaptic accumulator (float4, L2-resident 4MB buffer) ----
__global__ void zero_f4(float4* __restrict__ p, int n4) {
    int i = blockIdx.x * blockDim.x + threadIdx.x;
    if (i < n4) p[i] = make_float4(0.f, 0.f, 0.f, 0.f);
}

// ---- Ring-buffer shift: new_buf[:, N:] = z_buf[:, :N*(D-1)] ----
// Fallback path: plain float4 copy through VGPRs.
__global__ void shift_buf(const float4* __restrict__ zbuf, float4* __restrict__ obuf) {
    const int row4 = NN * (DD - 1) / 4;           // 50000 float4 per batch row
    int i = blockIdx.x * blockDim.x + threadIdx.x;
    if (i >= BB * row4) return;
    int b = i / row4, k = i - b * row4;
    obuf[(b * NN * DD + NN) / 4 + k] = zbuf[(b * NN * DD) / 4 + k];
}

// CDNA5 async path: global -> LDS -> global with ASYNCcnt-tracked DMA ops,
// zero VGPR data movement. Each lane owns a private 16B LDS slot, so a
// wave-level s_wait_asynccnt between load and store suffices (no barrier).
// Builtin signature (from compiler diagnostic): takes v4i* operands —
// global side in AS(1), LDS side in AS(3).
#if __has_builtin(__builtin_amdgcn_global_load_async_to_lds_b128) && \
    __has_builtin(__builtin_amdgcn_global_store_async_from_lds_b128)
#define HAVE_ASYNC_LDS 1
typedef int v4i __attribute__((ext_vector_type(4)));
typedef __attribute__((address_space(1))) v4i gv4i;   // global 16B chunk
typedef __attribute__((address_space(3))) v4i lv4i;   // LDS 16B chunk

__device__ __forceinline__ void wait_asynccnt0() {
#if __has_builtin(__builtin_amdgcn_s_wait_asynccnt)
    __builtin_amdgcn_s_wait_asynccnt(0);
#else
    asm volatile("s_wait_asynccnt 0x0" ::: "memory");
#endif
}

__global__ void shift_buf_async(const float* __restrict__ zbuf,
                                float* __restrict__ obuf) {
    __shared__ __align__(16) char lds[256 * 16];
    const int row16 = NN * (DD - 1) * 4 / 16;     // 50000 16B chunks per batch
    int i = blockIdx.x * blockDim.x + threadIdx.x;
    if (i >= BB * row16) return;
    int b = i / row16, k = i - b * row16;

    const char* src = (const char*)zbuf + (size_t)b * NN * DD * 4 + (size_t)k * 16;
    char*       dst = (char*)obuf + (size_t)b * NN * DD * 4 + (size_t)NN * 4
                                  + (size_t)k * 16;
    lv4i* l = (lv4i*)(void*)&lds[threadIdx.x * 16];

    __builtin_amdgcn_global_load_async_to_lds_b128(
        (gv4i*)(void*)src, l, /*offset=*/0, /*cpol=*/0);
    wait_asynccnt0();                              // LDS slot now holds the data
    __builtin_amdgcn_global_store_async_from_lds_b128(
        (gv4i*)(void*)dst, l, /*offset=*/0, /*cpol=*/0);
    // kernel-end implicit wait drains the outstanding async store
}
#endif

// ---- Edge scatter: acc[b, post, :] += (src[b, pre] * w) * factors[:]
// One thread per edge. LAZY loads: gather the 4 presynaptic activities first
// and early-exit when all are zero (92% of rec edges) so post/w/factors
// (28 B/edge) are only streamed for active edges. Exact: skipping +0.0 adds.
__global__ void edge_scatter(const float* __restrict__ src, int src_stride,
                             const int* __restrict__ pre,
                             const int* __restrict__ post,
                             const float* __restrict__ w,
                             const float* __restrict__ factors,
                             int E, float* __restrict__ acc) {
    int e = blockIdx.x * blockDim.x + threadIdx.x;
    if (e >= E) return;
    // Stream-ahead hint on the always-touched, fully sequential pre stream.
    __builtin_prefetch(&pre[e + 8 * 256], 0, 0);   // global_prefetch_b8

    int p = pre[e];
    float z[BB];
    bool any = false;
#pragma unroll
    for (int b = 0; b < BB; ++b) {
        z[b] = src[(size_t)b * src_stride + p];
        any = any || (z[b] != 0.0f);
    }
    if (!any) return;                              // ~92% of rec edges exit here

    int q = post[e];
    float wt = w[e];
    float f[NB];
#pragma unroll
    for (int k = 0; k < NB; ++k) f[k] = factors[(size_t)e * NB + k];

#pragma unroll
    for (int b = 0; b < BB; ++b) {
        if (z[b] != 0.0f) {
            float c = z[b] * wt;
            float* dst = acc + ((size_t)b * NN + q) * NB;
#pragma unroll
            for (int k = 0; k < NB; ++k) atomic_add_f32(dst + k, c * f[k]);
        }
    }
}

// ---- Per-neuron GLIF update ----
__global__ void neuron_update(
    const float* __restrict__ z_buf, const float* __restrict__ v,
    const float* __restrict__ r,     const float* __restrict__ asc1,
    const float* __restrict__ asc2,  const float* __restrict__ psc_rise,
    const float* __restrict__ psc,   const float* __restrict__ decay,
    const float* __restrict__ cf,    const float* __restrict__ v_th,
    const float* __restrict__ v_reset, const float* __restrict__ e_l,
    const float* __restrict__ g,     const float* __restrict__ t_ref,
    const float* __restrict__ asc_amps, const float* __restrict__ exp_dt_k,
    const float* __restrict__ syn_decay, const float* __restrict__ psc_initial,
    const float* __restrict__ acc,
    float* __restrict__ o_z,  float* __restrict__ o_v, float* __restrict__ o_r,
    float* __restrict__ o_a1, float* __restrict__ o_a2,
    float* __restrict__ o_pr, float* __restrict__ o_p, float* __restrict__ o_buf)
{
    int i = blockIdx.x * blockDim.x + threadIdx.x;
    if (i >= BB * NN) return;
    int b = i / NN, n = i - b * NN;

    float prev_z = z_buf[(size_t)b * NN * DD + n];
    size_t sb = (size_t)i * NB;   // == (b*NN + n)*NB : same layout for psc/acc

    float input_current = 0.f;
#pragma unroll
    for (int k = 0; k < NB; ++k) {
        float sd = syn_decay[k];       // uniform -> scalar loads
        float pi = psc_initial[k];
        float pr = psc_rise[sb + k];
        float pc = psc[sb + k];
        float si = acc[sb + k];
        float npr = sd * pr + pi * si;
        float np  = pc * sd + DT_C * sd * pr;
        input_current += np;
        o_pr[sb + k] = npr;
        o_p [sb + k] = np;
    }

    float a1 = asc1[i], a2 = asc2[i];
    float na1 = exp_dt_k[2 * n + 0] * a1 + prev_z * asc_amps[2 * n + 0];
    float na2 = exp_dt_k[2 * n + 1] * a2 + prev_z * asc_amps[2 * n + 1];

    float el = e_l[n];
    float vt = v_th[n];
    float c1 = input_current + a1 + a2 + g[n] * el;
    float nv = decay[n] * v[i] + cf[n] * c1;
    if (prev_z > 0.f) nv = v_reset[n];

    float nr  = fmaxf(r[i] + prev_z * t_ref[n] - DT_C, 0.f);
    float vsc = (nv - vt) / (vt - el + 1e-16f);
    float nz  = (vsc > 0.f) ? 1.f : 0.f;
    if (nr > 0.f) nz = 0.f;

    o_z[i] = nz; o_v[i] = nv; o_r[i] = nr; o_a1[i] = na1; o_a2[i] = na2;
    o_buf[(size_t)b * NN * DD + n] = nz;   // head of ring buffer
}

extern "C" void kernel_launch(void* const* d_in, const int* in_sizes, int n_in,
                              void* d_out, int out_size, void* d_ws, size_t ws_size,
                              hipStream_t stream) {
    // Inputs in setup_inputs() order
    const float* z_buf      = (const float*)d_in[0];
    const float* v          = (const float*)d_in[1];
    const float* r          = (const float*)d_in[2];
    const float* asc_1      = (const float*)d_in[3];
    const float* asc_2      = (const float*)d_in[4];
    const float* psc_rise   = (const float*)d_in[5];
    const float* psc        = (const float*)d_in[6];
    const float* lgn_x      = (const float*)d_in[7];
    const float* bkg_x      = (const float*)d_in[8];
    const float* rec_w      = (const float*)d_in[9];
    const float* rec_fac    = (const float*)d_in[10];
    const float* lgn_w      = (const float*)d_in[11];
    const float* lgn_fac    = (const float*)d_in[12];
    const float* bkg_w      = (const float*)d_in[13];
    const float* bkg_fac    = (const float*)d_in[14];
    const float* decay      = (const float*)d_in[15];
    const float* cur_fac    = (const float*)d_in[16];
    const float* v_th       = (const float*)d_in[17];
    const float* v_reset    = (const float*)d_in[18];
    const float* e_l        = (const float*)d_in[19];
    const float* g          = (const float*)d_in[20];
    const float* t_ref      = (const float*)d_in[21];
    const float* asc_amps   = (const float*)d_in[22];
    const float* exp_dt_k   = (const float*)d_in[23];
    const float* syn_decay  = (const float*)d_in[24];
    const float* psc_init   = (const float*)d_in[25];
    const int*   rec_pre    = (const int*)d_in[26];
    const int*   rec_post   = (const int*)d_in[27];
    const int*   lgn_pre    = (const int*)d_in[28];
    const int*   lgn_post   = (const int*)d_in[29];
    const int*   bkg_pre    = (const int*)d_in[30];
    const int*   bkg_post   = (const int*)d_in[31];

    float* out = (float*)d_out;
    float* acc = (float*)d_ws;                    // B*N*NB f32 = 4 MB (L2-resident)

    const int BN   = BB * NN;                     // 200000
    const int BNNB = BB * NN * NB;                // 1000000

    // Output section offsets (flat concat in return order)
    float* o_z   = out;
    float* o_v   = out + (size_t)1 * BN;
    float* o_r   = out + (size_t)2 * BN;
    float* o_a1  = out + (size_t)3 * BN;
    float* o_a2  = out + (size_t)4 * BN;
    float* o_pr  = out + (size_t)5 * BN;
    float* o_p   = out + (size_t)5 * BN + BNNB;
    float* o_buf = out + (size_t)5 * BN + 2 * (size_t)BNNB;

    const int TB = 256;                           // 8 wave32 per block

    // 1) zero accumulator (must be fresh every call: ws is not re-poisoned)
    int n4 = BNNB / 4;
    zero_f4<<<(n4 + TB - 1) / TB, TB, 0, stream>>>((float4*)acc, n4);

    // 2) three sparse scatter passes (rec gathers from z_buf rows of length N*D)
    edge_scatter<<<(E_REC + TB - 1) / TB, TB, 0, stream>>>(
        z_buf, NN * DD, rec_pre, rec_post, rec_w, rec_fac, E_REC, acc);
    edge_scatter<<<(E_LGN + TB - 1) / TB, TB, 0, stream>>>(
        lgn_x, N_LGN, lgn_pre, lgn_post, lgn_w, lgn_fac, E_LGN, acc);
    edge_scatter<<<(E_BKG + TB - 1) / TB, TB, 0, stream>>>(
        bkg_x, N_BKG, bkg_pre, bkg_post, bkg_w, bkg_fac, E_BKG, acc);

    // 3) per-neuron GLIF update (writes all point outputs + ring-buffer head)
    neuron_update<<<(BN + TB - 1) / TB, TB, 0, stream>>>(
        z_buf, v, r, asc_1, asc_2, psc_rise, psc, decay, cur_fac, v_th, v_reset,
        e_l, g, t_ref, asc_amps, exp_dt_k, syn_decay, psc_init, acc,
        o_z, o_v, o_r, o_a1, o_a2, o_pr, o_p, o_buf);

    // 4) ring-buffer shift: new_buf[:, N:] = z_buf[:, :N*(D-1)]
#if defined(HAVE_ASYNC_LDS)
    int nshift16 = BB * (NN * (DD - 1) / 4);      // 16B chunks
    shift_buf_async<<<(nshift16 + TB - 1) / TB, TB, 0, stream>>>(z_buf, o_buf);
#else
    int nshift = BB * (NN * (DD - 1) / 4);
    shift_buf<<<(nshift + TB - 1) / TB, TB, 0, stream>>>(
        (const float4*)z_buf, (float4*)o_buf);
#endif
}